// OCSVMguidedAutoencoder_68521908240667
// MI455X (gfx1250) — compile-verified
//
#include <hip/hip_runtime.h>
#include <hip/hip_bf16.h>
#include <math.h>

// ---------------------------------------------------------------------------
// OCSVM-guided autoencoder for MI455X (gfx1250, wave32, WMMA).
// Heavy path: 2048x2048 PSD matrix sqrt via Newton-Schulz (GEMM-only) using
// V_WMMA_F32_16X16X4_F32, with double-buffered async global->LDS staging
// (GLOBAL_LOAD_ASYNC_TO_LDS_B128 + s_wait_asynccnt) when available.
// CNN path: fused bandwidth-bound VALU kernels.
// ---------------------------------------------------------------------------

typedef float v2f __attribute__((ext_vector_type(2)));
typedef float v8f __attribute__((ext_vector_type(8)));

#define BN_EPS      0.001f
#define LRELU_A     0.3f

#if __has_builtin(__builtin_amdgcn_global_load_async_to_lds_b128) && \
    __has_builtin(__builtin_amdgcn_s_wait_asynccnt)
  #define USE_ASYNC_LDS 1
#else
  #define USE_ASYNC_LDS 0
#endif

#if USE_ASYNC_LDS
// Parameter types per hipcc diagnostic:
//   (int __vector_size(16) addrspace(1)*, int __vector_size(16) addrspace(3)*,
//    imm int offset, imm int cpol)
typedef int v4i_b128 __attribute__((vector_size(16)));
typedef __attribute__((address_space(1))) v4i_b128 gl_b128;
typedef __attribute__((address_space(3))) v4i_b128 ls_b128;

__device__ __forceinline__ void async_ld_b128(const void* g, void* l) {
  __builtin_amdgcn_global_load_async_to_lds_b128((gl_b128*)g, (ls_b128*)l, 0, 0);
}
#endif

// ======================= WMMA f32 GEMM (NN, row-major) =====================
// C[M,N] = alpha * A[M,K] @ B[K,N] + beta * Cin[M,N]
// Block: 256 threads (8 waves, 4x2). Block tile 128x64, wave tile 32x32
// (2x2 fragments of 16x16), K staged in LDS chunks of 16, double-buffered.
// Staging assumes full tiles: M % 128 == 0, N % 64 == 0, K % 16 == 0
// (true for every call here: M=N=2048, K in {32, 2048}).

#define GBM 128
#define GBN 64
#define GBK 16
#define APAD 4   // 16B-aligned row stride, bank-conflict-free fragment reads

__device__ __forceinline__ v8f wmma_f32_16x16x4(v2f a, v2f b, v8f c) {
  // 8 args: (neg_a, A, neg_b, B, c_mod, C, reuse_a, reuse_b)
  return __builtin_amdgcn_wmma_f32_16x16x4_f32(false, a, false, b, (short)0, c,
                                               false, false);
}

__global__ __launch_bounds__(256)
void wmma_gemm_nn(const float* __restrict__ A, const float* __restrict__ B,
                  const float* __restrict__ Cin, float* __restrict__ C,
                  int M, int N, int K, float alpha, float beta)
{
  __shared__ float As[2][GBM][GBK + APAD];
  __shared__ float Bs[2][GBK][GBN + APAD];

  const int tid   = threadIdx.x;
  const int wave  = tid >> 5;          // 0..7
  const int lane  = tid & 31;
  const int hf    = lane >> 4;         // 0/1 (wave half per ISA layouts)
  const int lr    = lane & 15;
  const int waveM = wave >> 1;         // 0..3 -> 32-row band
  const int waveN = wave & 1;          // 0..1 -> 32-col band

  const int blockRow = blockIdx.y * GBM;
  const int blockCol = blockIdx.x * GBN;

  // Stage one K-tile (16B chunks; 3 vmem instructions per wave total).
  auto stage_tile = [&](int buf, int k0) {
    #pragma unroll
    for (int p = 0; p < 2; ++p) {           // A: 128x16 fl = 512 chunks
      int q  = tid + p * 256;
      int r  = q >> 2;
      int c4 = (q & 3) * 4;
      const float* g = A + (size_t)(blockRow + r) * K + k0 + c4;
      float*       l = &As[buf][r][c4];
#if USE_ASYNC_LDS
      async_ld_b128(g, l);
#else
      *(float4*)l = *(const float4*)g;
#endif
    }
    {                                        // B: 16x64 fl = 256 chunks
      int r  = tid >> 4;
      int c4 = (tid & 15) * 4;
      const float* g = B + (size_t)(k0 + r) * N + blockCol + c4;
      float*       l = &Bs[buf][r][c4];
#if USE_ASYNC_LDS
      async_ld_b128(g, l);
#else
      *(float4*)l = *(const float4*)g;
#endif
    }
  };

  v8f acc[2][2] = {};                  // 16x16 f32 accumulators
  const int nk = K / GBK;

  stage_tile(0, 0);
  int buf = 0;
  for (int kt = 0; kt < nk; ++kt) {
#if USE_ASYNC_LDS
    __builtin_amdgcn_s_wait_asynccnt(0);   // this stage's LDS data landed
#endif
    __syncthreads();                       // all waves' data visible; prev
                                           // compute on buf^1 fully retired
    if (kt + 1 < nk) stage_tile(buf ^ 1, (kt + 1) * GBK);

    #pragma unroll
    for (int kk = 0; kk < GBK; kk += 4) {
      // A fragment (16x4 f32): lane lr holds row lr, k = 2*hf + {0,1}
      v2f af[2];
      #pragma unroll
      for (int i = 0; i < 2; ++i) {
        int row = waveM * 32 + i * 16 + lr;
        af[i][0] = As[buf][row][kk + 2 * hf + 0];
        af[i][1] = As[buf][row][kk + 2 * hf + 1];
      }
      // B fragment (4x16 f32): lane lr holds col lr, k = 2*hf + {0,1}
      v2f bf[2];
      #pragma unroll
      for (int j = 0; j < 2; ++j) {
        int col = waveN * 32 + j * 16 + lr;
        bf[j][0] = Bs[buf][kk + 2 * hf + 0][col];
        bf[j][1] = Bs[buf][kk + 2 * hf + 1][col];
      }
      #pragma unroll
      for (int i = 0; i < 2; ++i)
        #pragma unroll
        for (int j = 0; j < 2; ++j)
          acc[i][j] = wmma_f32_16x16x4(af[i], bf[j], acc[i][j]);
    }
    buf ^= 1;
  }

  // ---- epilogue: C = alpha*acc + beta*Cin ----
  // C/D layout: VGPR v holds M = v + 8*hf, N = lr
  #pragma unroll
  for (int i = 0; i < 2; ++i) {
    #pragma unroll
    for (int j = 0; j < 2; ++j) {
      int row0 = blockRow + waveM * 32 + i * 16;
      int col  = blockCol + waveN * 32 + j * 16 + lr;
      #pragma unroll
      for (int v = 0; v < 8; ++v) {
        int row = row0 + v + 8 * hf;
        if (row < M && col < N) {
          float val = alpha * acc[i][j][v];
          if (beta != 0.f) val += beta * Cin[(size_t)row * N + col];
          C[(size_t)row * N + col] = val;
        }
      }
    }
  }
}

// =========================== CNN kernels ===================================
// MODE 0: bias + BN + leaky-relu ; MODE 1: bias + sigmoid
template<int CIN, int COUT, int MODE>
__global__ __launch_bounds__(256)
void conv5x5(const float* __restrict__ x, const float* __restrict__ w,
             const float* __restrict__ bias,
             const float* __restrict__ bng, const float* __restrict__ bnb,
             const float* __restrict__ bnm, const float* __restrict__ bnv,
             float* __restrict__ y, int B, int H, int W)
{
  int idx = blockIdx.x * blockDim.x + threadIdx.x;
  int total = B * H * W;
  if (idx >= total) return;
  int xw = idx % W; int t = idx / W; int yh = t % H; int b = t / H;

  float acc[COUT];
  #pragma unroll
  for (int co = 0; co < COUT; ++co) acc[co] = bias[co];

  #pragma unroll
  for (int ky = 0; ky < 5; ++ky) {
    int iy = yh + ky - 2;
    if (iy < 0 || iy >= H) continue;
    #pragma unroll
    for (int kx = 0; kx < 5; ++kx) {
      int ix = xw + kx - 2;
      if (ix < 0 || ix >= W) continue;
      const float* xp = x + ((size_t)(b * H + iy) * W + ix) * CIN;
      const float* wp = w + (size_t)((ky * 5 + kx) * CIN) * COUT;
      #pragma unroll
      for (int ci = 0; ci < CIN; ++ci) {
        float xv = xp[ci];
        #pragma unroll
        for (int co = 0; co < COUT; ++co)
          acc[co] = fmaf(xv, wp[ci * COUT + co], acc[co]);
      }
    }
  }

  float* yp = y + (size_t)idx * COUT;
  #pragma unroll
  for (int co = 0; co < COUT; ++co) {
    float v = acc[co];
    if (MODE == 0) {
      v = (v - bnm[co]) * rsqrtf(bnv[co] + BN_EPS) * bng[co] + bnb[co];
      v = (v >= 0.f) ? v : LRELU_A * v;
    } else {
      v = 1.f / (1.f + expf(-v));
    }
    yp[co] = v;
  }
}

template<int C>
__global__ __launch_bounds__(256)
void maxpool2x2(const float* __restrict__ x, float* __restrict__ y,
                int B, int H, int W)
{
  int OH = H >> 1, OW = W >> 1;
  int idx = blockIdx.x * blockDim.x + threadIdx.x;
  int total = B * OH * OW;
  if (idx >= total) return;
  int ox = idx % OW; int t = idx / OW; int oy = t % OH; int b = t / OH;
  const float* p0 = x + ((size_t)(b * H + oy * 2) * W + ox * 2) * C;
  const float* p1 = p0 + (size_t)W * C;
  float* yp = y + (size_t)idx * C;
  #pragma unroll
  for (int c = 0; c < C; ++c)
    yp[c] = fmaxf(fmaxf(p0[c], p0[C + c]), fmaxf(p1[c], p1[C + c]));
}

template<int C>
__global__ __launch_bounds__(256)
void upsample2x2(const float* __restrict__ x, float* __restrict__ y,
                 int B, int H, int W)
{
  int OH = H * 2, OW = W * 2;
  int idx = blockIdx.x * blockDim.x + threadIdx.x;
  int total = B * OH * OW;
  if (idx >= total) return;
  int ox = idx % OW; int t = idx / OW; int oy = t % OH; int b = t / OH;
  const float* xp = x + ((size_t)(b * H + (oy >> 1)) * W + (ox >> 1)) * C;
  float* yp = y + (size_t)idx * C;
  #pragma unroll
  for (int c = 0; c < C; ++c) yp[c] = xp[c];
}

__global__ __launch_bounds__(256)
void dense_bias(const float* __restrict__ X, const float* __restrict__ Wm,
                const float* __restrict__ bias, float* __restrict__ Y,
                int Bn, int K, int N)
{
  int idx = blockIdx.x * blockDim.x + threadIdx.x;
  if (idx >= Bn * N) return;
  int n = idx % N, b = idx / N;
  float s = bias[n];
  const float* xr = X + (size_t)b * K;
  for (int k = 0; k < K; ++k) s = fmaf(xr[k], Wm[(size_t)k * N + n], s);
  Y[idx] = s;
}

// ======================= OCSVM / kernel-matrix path ========================

__global__ __launch_bounds__(256)
void standardize_cols(const float* __restrict__ latent, float* __restrict__ Zs,
                      int n, int d)
{
  __shared__ float s1[256], s2[256];
  int j = blockIdx.x, tid = threadIdx.x;
  float sum = 0.f, sq = 0.f;
  for (int i = tid; i < n; i += 256) {
    float v = latent[(size_t)i * d + j];
    sum += v; sq += v * v;
  }
  s1[tid] = sum; s2[tid] = sq; __syncthreads();
  for (int s = 128; s > 0; s >>= 1) {
    if (tid < s) { s1[tid] += s1[tid + s]; s2[tid] += s2[tid + s]; }
    __syncthreads();
  }
  float mean = s1[0] / n;
  float var  = s2[0] / n - mean * mean;
  float inv  = rsqrtf(fmaxf(var, 1e-30f));
  for (int i = tid; i < n; i += 256)
    Zs[(size_t)i * d + j] = (latent[(size_t)i * d + j] - mean) * inv;
}

// scal[0]=gamma, scal[1]=eps
__global__ __launch_bounds__(256)
void gamma_kernel(const float* __restrict__ Zs, float* __restrict__ scal,
                  int count, float zdim)
{
  __shared__ float s1[256], s2[256];
  int tid = threadIdx.x;
  float sum = 0.f, sq = 0.f;
  for (int i = tid; i < count; i += 256) { float v = Zs[i]; sum += v; sq += v * v; }
  s1[tid] = sum; s2[tid] = sq; __syncthreads();
  for (int s = 128; s > 0; s >>= 1) {
    if (tid < s) { s1[tid] += s1[tid + s]; s2[tid] += s2[tid + s]; }
    __syncthreads();
  }
  if (tid == 0) {
    float mean = s1[0] / count;
    float var  = s2[0] / count - mean * mean;
    float gamma = 1.f / (zdim * var);
    scal[0] = gamma;
    scal[1] = 1e-8f / gamma;
  }
}

__global__ __launch_bounds__(256)
void transpose_kernel(const float* __restrict__ A, float* __restrict__ At,
                      int n, int d)
{
  int idx = blockIdx.x * blockDim.x + threadIdx.x;
  if (idx >= n * d) return;
  int j = idx % d, i = idx / d;
  At[(size_t)j * n + i] = A[idx];
}

// k = exp(-gamma*(Gii+Gjj-2Gij)); kout = k (d_out), kreg = k + eps*I (ws)
__global__ __launch_bounds__(256)
void rbf_kernel(const float* __restrict__ G, const float* __restrict__ scal,
                float* __restrict__ kout, float* __restrict__ kreg, int n)
{
  size_t idx = (size_t)blockIdx.x * blockDim.x + threadIdx.x;
  if (idx >= (size_t)n * n) return;
  int i = (int)(idx / n), j = (int)(idx % n);
  float d2 = G[(size_t)i * n + i] + G[(size_t)j * n + j] - 2.f * G[idx];
  d2 = fmaxf(d2, 0.f);
  float kv = expf(-scal[0] * d2);
  kout[idx] = kv;
  kreg[idx] = (i == j) ? kv + scal[1] : kv;
}

__global__ __launch_bounds__(256)
void sumsq_partial(const float* __restrict__ A, float* __restrict__ partial,
                   int count)
{
  __shared__ float s1[256];
  int tid = threadIdx.x;
  float s = 0.f;
  for (int i = blockIdx.x * 256 + tid; i < count; i += gridDim.x * 256) {
    float v = A[i]; s += v * v;
  }
  s1[tid] = s; __syncthreads();
  for (int k = 128; k > 0; k >>= 1) {
    if (tid < k) s1[tid] += s1[tid + k];
    __syncthreads();
  }
  if (tid == 0) partial[blockIdx.x] = s1[0];
}

// scal[2]=||A||_F, scal[3]=1/||A||_F, scal[4]=sqrt(||A||_F)
__global__ __launch_bounds__(256)
void finalize_norm(const float* __restrict__ partial, int nparts,
                   float* __restrict__ scal)
{
  __shared__ float s1[256];
  int tid = threadIdx.x;
  float s = 0.f;
  for (int i = tid; i < nparts; i += 256) s += partial[i];
  s1[tid] = s; __syncthreads();
  for (int k = 128; k > 0; k >>= 1) {
    if (tid < k) s1[tid] += s1[tid + k];
    __syncthreads();
  }
  if (tid == 0) {
    float fro = sqrtf(s1[0]);
    scal[2] = fro; scal[3] = 1.f / fro; scal[4] = sqrtf(fro);
  }
}

// Y0 = Kreg/||Kreg||_F ; Z0 = I
__global__ __launch_bounds__(256)
void ns_init(const float* __restrict__ Kreg, const float* __restrict__ scal,
             float* __restrict__ Y, float* __restrict__ Z, int n)
{
  size_t idx = (size_t)blockIdx.x * blockDim.x + threadIdx.x;
  if (idx >= (size_t)n * n) return;
  int i = (int)(idx / n), j = (int)(idx % n);
  Y[idx] = Kreg[idx] * scal[3];
  Z[idx] = (i == j) ? 1.f : 0.f;
}

// k_sqrt = sqrt(||Kreg||_F) * Y
__global__ __launch_bounds__(256)
void scale_out(const float* __restrict__ Y, const float* __restrict__ scal,
               float* __restrict__ out, int count)
{
  int idx = blockIdx.x * blockDim.x + threadIdx.x;
  if (idx >= count) return;
  out[idx] = Y[idx] * scal[4];
}

// ================================ launch ===================================

static inline int cdiv(int a, int b) { return (a + b - 1) / b; }

extern "C" void kernel_launch(void* const* d_in, const int* in_sizes, int n_in,
                              void* d_out, int out_size, void* d_ws, size_t ws_size,
                              hipStream_t stream)
{
  const float* x    = (const float*)d_in[0];
  const float* c1w  = (const float*)d_in[1];
  const float* c1b  = (const float*)d_in[2];
  const float* bn1g = (const float*)d_in[3];
  const float* bn1b = (const float*)d_in[4];
  const float* bn1m = (const float*)d_in[5];
  const float* bn1v = (const float*)d_in[6];
  const float* c2w  = (const float*)d_in[7];
  const float* c2b  = (const float*)d_in[8];
  const float* bn2g = (const float*)d_in[9];
  const float* bn2b = (const float*)d_in[10];
  const float* bn2m = (const float*)d_in[11];
  const float* bn2v = (const float*)d_in[12];
  const float* edw  = (const float*)d_in[13];
  const float* edb  = (const float*)d_in[14];
  const float* ddw  = (const float*)d_in[15];
  const float* ddb  = (const float*)d_in[16];
  const float* t1w  = (const float*)d_in[17];
  const float* t1b  = (const float*)d_in[18];
  const float* dbn1g = (const float*)d_in[19];
  const float* dbn1b = (const float*)d_in[20];
  const float* dbn1m = (const float*)d_in[21];
  const float* dbn1v = (const float*)d_in[22];
  const float* t2w  = (const float*)d_in[23];
  const float* t2b  = (const float*)d_in[24];
  const float* dbn2g = (const float*)d_in[25];
  const float* dbn2b = (const float*)d_in[26];
  const float* dbn2m = (const float*)d_in[27];
  const float* dbn2v = (const float*)d_in[28];
  const float* ow   = (const float*)d_in[29];
  const float* ob   = (const float*)d_in[30];

  const int B = in_sizes[0] / (28 * 28);   // 4096
  const int NH = B / 2;                    // 2048 rows of z_sv
  const int ZD = 32;                       // latent dim

  // ---- output layout (flat, in return order) ----
  float* out = (float*)d_out;
  float* out_decoded = out;                                   // B*28*28*1
  float* out_latent  = out + (size_t)B * 28 * 28;             // B*32
  float* out_ksv     = out_latent + (size_t)B * ZD;           // NH*NH
  float* out_ksqrt   = out_ksv + (size_t)NH * NH;             // NH*NH

  // ---- workspace arena (floats, 256B-aligned chunks) ----
  float* ws = (float*)d_ws;
  size_t off = 0;
  auto alloc = [&](size_t n) { float* p = ws + off; off += (n + 63) & ~(size_t)63; return p; };

  float* scal = alloc(16);                    // [gamma, eps, fro, 1/fro, sqrt(fro)]
  float* part = alloc(1024);
  float* Zs   = alloc((size_t)NH * ZD);
  float* ZsT  = alloc((size_t)NH * ZD);
  float* G    = alloc((size_t)NH * NH);
  float* Kreg = alloc((size_t)NH * NH);
  float* Ybuf = alloc((size_t)NH * NH);
  float* Zbuf = alloc((size_t)NH * NH);
  float* Tbuf = alloc((size_t)NH * NH);
  float* Y2   = alloc((size_t)NH * NH);
  float* Z2   = alloc((size_t)NH * NH);
  // CNN activation regions (reused across lifetimes)
  float* R0 = alloc((size_t)B * 28 * 28 * 8);   // h1 (B,28,28,4) then u2 (B,28,28,8)
  float* R1 = alloc((size_t)B * 14 * 14 * 8);   // p1 (B,14,14,4) then u1 (B,14,14,8)
  float* R2 = alloc((size_t)B * 14 * 14 * 8);   // h2 (B,14,14,8) then t1 (B,14,14,8)
  float* R3 = alloc((size_t)B * 7 * 7 * 8);     // p2 (B,7,7,8)  then dd (B,392)
  float* R4 = alloc((size_t)B * 28 * 28 * 4);   // t2 (B,28,28,4)
  (void)ws_size; (void)out_size; (void)n_in;

  const dim3 TB(256);

  // ============================ encoder ============================
  conv5x5<1, 4, 0><<<cdiv(B * 28 * 28, 256), TB, 0, stream>>>(
      x, c1w, c1b, bn1g, bn1b, bn1m, bn1v, R0, B, 28, 28);
  maxpool2x2<4><<<cdiv(B * 14 * 14, 256), TB, 0, stream>>>(R0, R1, B, 28, 28);
  conv5x5<4, 8, 0><<<cdiv(B * 14 * 14, 256), TB, 0, stream>>>(
      R1, c2w, c2b, bn2g, bn2b, bn2m, bn2v, R2, B, 14, 14);
  maxpool2x2<8><<<cdiv(B * 7 * 7, 256), TB, 0, stream>>>(R2, R3, B, 14, 14);
  dense_bias<<<cdiv(B * ZD, 256), TB, 0, stream>>>(R3, edw, edb, out_latent,
                                                   B, 392, ZD);

  // ============================ decoder ============================
  dense_bias<<<cdiv(B * 392, 256), TB, 0, stream>>>(out_latent, ddw, ddb, R3,
                                                    B, ZD, 392);
  upsample2x2<8><<<cdiv(B * 14 * 14, 256), TB, 0, stream>>>(R3, R1, B, 7, 7);
  // stride-1 SAME conv_transpose (5x5, symmetric pad) == SAME conv
  conv5x5<8, 8, 0><<<cdiv(B * 14 * 14, 256), TB, 0, stream>>>(
      R1, t1w, t1b, dbn1g, dbn1b, dbn1m, dbn1v, R2, B, 14, 14);
  upsample2x2<8><<<cdiv(B * 28 * 28, 256), TB, 0, stream>>>(R2, R0, B, 14, 14);
  conv5x5<8, 4, 0><<<cdiv(B * 28 * 28, 256), TB, 0, stream>>>(
      R0, t2w, t2b, dbn2g, dbn2b, dbn2m, dbn2v, R4, B, 28, 28);
  conv5x5<4, 1, 1><<<cdiv(B * 28 * 28, 256), TB, 0, stream>>>(
      R4, ow, ob, nullptr, nullptr, nullptr, nullptr, out_decoded, B, 28, 28);

  // ==================== RBF kernel matrix (WMMA) ===================
  standardize_cols<<<ZD, TB, 0, stream>>>(out_latent, Zs, NH, ZD);
  gamma_kernel<<<1, TB, 0, stream>>>(Zs, scal, NH * ZD, (float)ZD);
  transpose_kernel<<<cdiv(NH * ZD, 256), TB, 0, stream>>>(Zs, ZsT, NH, ZD);

  auto gemm = [&](const float* A, const float* Bm, const float* Cin, float* C,
                  int M, int N, int K, float alpha, float beta) {
    dim3 grid(cdiv(N, GBN), cdiv(M, GBM));
    wmma_gemm_nn<<<grid, TB, 0, stream>>>(A, Bm, Cin, C, M, N, K, alpha, beta);
  };

  // Gram: G = Zs @ Zs^T  (2048x2048, K=32) — row norms come from diag(G)
  gemm(Zs, ZsT, nullptr, G, NH, NH, ZD, 1.f, 0.f);
  rbf_kernel<<<cdiv(NH * NH, 256), TB, 0, stream>>>(G, scal, out_ksv, Kreg, NH);

  // ======== PSD matrix sqrt via Newton-Schulz (pure WMMA GEMMs) =====
  sumsq_partial<<<1024, TB, 0, stream>>>(Kreg, part, NH * NH);
  finalize_norm<<<1, TB, 0, stream>>>(part, 1024, scal);
  ns_init<<<cdiv(NH * NH, 256), TB, 0, stream>>>(Kreg, scal, Ybuf, Zbuf, NH);

  float* Yc = Ybuf; float* Zc = Zbuf; float* Yn = Y2; float* Zn = Z2;
  const int NS_ITERS = 20;
  for (int it = 0; it < NS_ITERS; ++it) {
    // T = Z @ Y
    gemm(Zc, Yc, nullptr, Tbuf, NH, NH, NH, 1.f, 0.f);
    // Y' = 1.5*Y - 0.5*(Y @ T)
    gemm(Yc, Tbuf, Yc, Yn, NH, NH, NH, -0.5f, 1.5f);
    // Z' = 1.5*Z - 0.5*(T @ Z)
    gemm(Tbuf, Zc, Zc, Zn, NH, NH, NH, -0.5f, 1.5f);
    float* t;
    t = Yc; Yc = Yn; Yn = t;
    t = Zc; Zc = Zn; Zn = t;
  }

  // k_z_sqrt = sqrt(||Kreg||_F) * Y_final
  scale_out<<<cdiv(NH * NH, 256), TB, 0, stream>>>(Yc, scal, out_ksqrt, NH * NH);
}